// Decoder_34694745817096
// MI455X (gfx1250) — compile-verified
//
#include <hip/hip_runtime.h>
#include <hip/hip_bf16.h>
#include <math.h>

#define NN 1024
#define NB 16
#define HD 256
#define D0 71
#define D0P 72
#define NPAIRS ((NN * (NN - 1)) / 2)

typedef __attribute__((ext_vector_type(2))) float v2f;
typedef __attribute__((ext_vector_type(8))) float v8f;

// ---------------------------------------------------------------------------
// Wave-level f32 WMMA tile:  D[16x16] += A[16xK] * B[Kx16]
//   A: LDS activations, row-major [16][actLd], zero-padded to Kp (mult of 4)
//   B[k][n] = W[n0+n][k]  (W row-major [Ncols][ldw]); guard k<K on the W side
// VGPR layouts per CDNA5 ISA 7.12.2 (32-bit A 16x4, 32-bit C/D 16x16):
//   A: vgpr0 = K=0 (lanes 0-15) / K=2 (lanes 16-31); vgpr1 = K=1 / K=3
//   B: mirrored (row K striped across 16 lanes)
//   D: vgpr r -> M = r + (lane<16 ? 0 : 8), N = lane&15
// ---------------------------------------------------------------------------
__device__ inline v8f wmma_tile_f32(const float* __restrict__ W, int ldw, int K,
                                    int Kp, const float* __restrict__ act,
                                    int actLd, int n0, int lane) {
  v8f acc = {};
  const int half = (lane >> 4) << 1;  // 0 for lanes 0-15, 2 for lanes 16-31
  const int m = lane & 15;
  const float* wrow = W + (size_t)(n0 + m) * (size_t)ldw;
  for (int k0 = 0; k0 < Kp; k0 += 4) {
    const int ka = k0 + half;
    v2f a, b;
    a.x = act[m * actLd + ka];
    a.y = act[m * actLd + ka + 1];
    b.x = (ka < K) ? wrow[ka] : 0.0f;
    b.y = (ka + 1 < K) ? wrow[ka + 1] : 0.0f;
    acc = __builtin_amdgcn_wmma_f32_16x16x4_f32(false, a, false, b, (short)0,
                                                acc, false, false);
  }
  return acc;
}

__device__ inline float gelu_exact(float x) {
  return 0.5f * x * (1.0f + erff(x * 0.70710678118654752f));
}

__device__ inline void ln_row(const float* __restrict__ in, float* __restrict__ out,
                              const float* __restrict__ g, const float* __restrict__ b,
                              int K) {
  float m = 0.0f;
  for (int k = 0; k < K; ++k) m += in[k];
  m /= (float)K;
  float v = 0.0f;
  for (int k = 0; k < K; ++k) { float d = in[k] - m; v += d * d; }
  v /= (float)K;
  const float inv = 1.0f / sqrtf(v + 1e-5f);
  for (int k = 0; k < K; ++k) out[k] = (in[k] - m) * inv * g[k] + b[k];
}

// ---------------------------------------------------------------------------
// Phase A: the whole network collapses to per-batch [16 x 256] vectors
// (broadcast nodes => uniform attention => MHA == V projection).
// One block, 16 waves, each wave owns one 16-column WMMA tile.
// Produces logits_ws[16][2].
// ---------------------------------------------------------------------------
__global__ __launch_bounds__(512) void decoder_phaseA(
    const float* __restrict__ x, const float* __restrict__ stats,
    const float* __restrict__ ln0_g, const float* __restrict__ ln0_b,
    const float* __restrict__ rb1_ln_g, const float* __restrict__ rb1_ln_b,
    const float* __restrict__ rb1_w1, const float* __restrict__ rb1_b1,
    const float* __restrict__ rb1_w2, const float* __restrict__ rb1_b2,
    const float* __restrict__ rb1_wp, const float* __restrict__ rb1_bp,
    const float* __restrict__ rb2_ln_g, const float* __restrict__ rb2_ln_b,
    const float* __restrict__ rb2_w1, const float* __restrict__ rb2_b1,
    const float* __restrict__ rb2_w2, const float* __restrict__ rb2_b2,
    const float* __restrict__ att_ln_g, const float* __restrict__ att_ln_b,
    const float* __restrict__ att_win, const float* __restrict__ att_bin,
    const float* __restrict__ att_wout, const float* __restrict__ att_bout,
    const float* __restrict__ out_w, const float* __restrict__ out_b,
    const float* __restrict__ fin_w, const float* __restrict__ fin_b,
    float* __restrict__ logits_ws) {
  __shared__ float sZ[NB][D0P];  // ln0(concat(x,stats)), zero-padded
  __shared__ float sT[NB][D0P];  // rb1_ln(sZ)
  __shared__ float sA[NB][HD];
  __shared__ float sV[NB][HD];
  __shared__ float sH[NB][HD];

  const int tid = threadIdx.x;
  const int lane = tid & 31;
  const int wave = tid >> 5;
  const int n0 = wave * 16;
  const int col = n0 + (lane & 15);
  const int r0 = (lane >> 4) * 8;

  // concat(x, stats) -> sZ (raw), zero pad col 71
  for (int idx = tid; idx < NB * D0P; idx += blockDim.x) {
    const int b = idx / D0P, k = idx % D0P;
    float v = 0.0f;
    if (k < 64) v = x[b * 64 + k];
    else if (k < D0) v = stats[b * 7 + (k - 64)];
    sZ[b][k] = v;
  }
  __syncthreads();

  // ln0 in-place, then rb1_ln -> sT
  if (tid < NB) {
    ln_row(&sZ[tid][0], &sZ[tid][0], ln0_g, ln0_b, D0);
    sZ[tid][D0] = 0.0f;
    ln_row(&sZ[tid][0], &sT[tid][0], rb1_ln_g, rb1_ln_b, D0);
    sT[tid][D0] = 0.0f;
  }
  __syncthreads();

  // rb1: sA = gelu(sT @ rb1_w1^T + b1)
  {
    v8f acc = wmma_tile_f32(rb1_w1, D0, D0, D0P, &sT[0][0], D0P, n0, lane);
    const float bb = rb1_b1[col];
    for (int r = 0; r < 8; ++r) sA[r0 + r][col] = gelu_exact(acc[r] + bb);
  }
  __syncthreads();

  // rb1: sH = sA @ rb1_w2^T + b2 + sZ @ rb1_wp^T + bp
  {
    v8f acc = wmma_tile_f32(rb1_w2, HD, HD, HD, &sA[0][0], HD, n0, lane);
    v8f accp = wmma_tile_f32(rb1_wp, D0, D0, D0P, &sZ[0][0], D0P, n0, lane);
    const float bb = rb1_b2[col] + rb1_bp[col];
    for (int r = 0; r < 8; ++r) sH[r0 + r][col] = acc[r] + accp[r] + bb;
  }
  __syncthreads();

  // rb2: sA = LN(sH)
  if (tid < NB) ln_row(&sH[tid][0], &sA[tid][0], rb2_ln_g, rb2_ln_b, HD);
  __syncthreads();

  // rb2: sV = gelu(sA @ rb2_w1^T + b1)
  {
    v8f acc = wmma_tile_f32(rb2_w1, HD, HD, HD, &sA[0][0], HD, n0, lane);
    const float bb = rb2_b1[col];
    for (int r = 0; r < 8; ++r) sV[r0 + r][col] = gelu_exact(acc[r] + bb);
  }
  __syncthreads();

  // rb2: sH += sV @ rb2_w2^T + b2
  {
    v8f acc = wmma_tile_f32(rb2_w2, HD, HD, HD, &sV[0][0], HD, n0, lane);
    const float bb = rb2_b2[col];
    for (int r = 0; r < 8; ++r) sH[r0 + r][col] += acc[r] + bb;
  }
  __syncthreads();

  // attention: uniform softmax over identical tokens => o == v.
  // sA = LN(sH, att); sV = sA @ Wv^T + bv  (Wv = att_win rows [2H,3H))
  if (tid < NB) ln_row(&sH[tid][0], &sA[tid][0], att_ln_g, att_ln_b, HD);
  __syncthreads();
  {
    v8f acc = wmma_tile_f32(att_win + (size_t)2 * HD * HD, HD, HD, HD,
                            &sA[0][0], HD, n0, lane);
    const float bb = att_bin[2 * HD + col];
    for (int r = 0; r < 8; ++r) sV[r0 + r][col] = acc[r] + bb;
  }
  __syncthreads();

  // sA = sV @ att_wout^T + bout
  {
    v8f acc = wmma_tile_f32(att_wout, HD, HD, HD, &sV[0][0], HD, n0, lane);
    const float bb = att_bout[col];
    for (int r = 0; r < 8; ++r) sA[r0 + r][col] = acc[r] + bb;
  }
  __syncthreads();

  // sV = sA @ out_w^T + out_b   (final h, identical for every node)
  {
    v8f acc = wmma_tile_f32(out_w, HD, HD, HD, &sA[0][0], HD, n0, lane);
    const float bb = out_b[col];
    for (int r = 0; r < 8; ++r) sV[r0 + r][col] = acc[r] + bb;
  }
  __syncthreads();

  // logits[b][c] = sum_k h[b][k] * (fin_w[c][k] + fin_w[c][H+k]) + fin_b[c]
  if (tid < 2 * NB) {
    const int b = tid >> 1, c = tid & 1;
    float s = fin_b[c];
    const float* fw = fin_w + (size_t)c * 2 * HD;
    for (int k = 0; k < HD; ++k) s += sV[b][k] * (fw[k] + fw[HD + k]);
    logits_ws[b * 2 + c] = s;
  }
}

// ---------------------------------------------------------------------------
// Phase B: streaming pass over all pairs. Hard gumbel-softmax forward value
// is one_hot(argmax), and argmax((l+g)/tau) == argmax(l+g) for tau>0.
// Writes adj[i][j] and adj[j][i]; diagonal zeroed. ~128 MB of HBM traffic.
// ---------------------------------------------------------------------------
__global__ __launch_bounds__(256) void decoder_pairs(
    const float* __restrict__ u, const float* __restrict__ logits_ws,
    float* __restrict__ adj) {
  const int i = blockIdx.x;
  const int b = blockIdx.y;
  const float l0 = logits_ws[b * 2 + 0];
  const float l1 = logits_ws[b * 2 + 1];
  float* adjB = adj + (size_t)b * NN * NN;
  if (threadIdx.x == 0) adjB[(size_t)i * NN + i] = 0.0f;
  const long rowOff = (long)i * (NN - 1) - ((long)i * (i - 1)) / 2;
  const float2* __restrict__ u2 = (const float2*)u;
  for (int j = i + 1 + (int)threadIdx.x; j < NN; j += (int)blockDim.x) {
    const long p = rowOff + (j - i - 1);
    const float2 uv = u2[(size_t)b * NPAIRS + p];
    const float g0 = -logf(-logf(uv.x + 1e-10f) + 1e-10f);
    const float g1 = -logf(-logf(uv.y + 1e-10f) + 1e-10f);
    const float e = (l0 + g0 >= l1 + g1) ? 1.0f : 0.0f;
    adjB[(size_t)i * NN + j] = e;
    adjB[(size_t)j * NN + i] = e;
  }
}

extern "C" void kernel_launch(void* const* d_in, const int* in_sizes, int n_in,
                              void* d_out, int out_size, void* d_ws, size_t ws_size,
                              hipStream_t stream) {
  const float* x        = (const float*)d_in[0];
  const float* stats    = (const float*)d_in[1];
  const float* u        = (const float*)d_in[2];
  const float* ln0_g    = (const float*)d_in[3];
  const float* ln0_b    = (const float*)d_in[4];
  const float* rb1_ln_g = (const float*)d_in[5];
  const float* rb1_ln_b = (const float*)d_in[6];
  const float* rb1_w1   = (const float*)d_in[7];
  const float* rb1_b1   = (const float*)d_in[8];
  const float* rb1_w2   = (const float*)d_in[9];
  const float* rb1_b2   = (const float*)d_in[10];
  const float* rb1_wp   = (const float*)d_in[11];
  const float* rb1_bp   = (const float*)d_in[12];
  const float* rb2_ln_g = (const float*)d_in[13];
  const float* rb2_ln_b = (const float*)d_in[14];
  const float* rb2_w1   = (const float*)d_in[15];
  const float* rb2_b1   = (const float*)d_in[16];
  const float* rb2_w2   = (const float*)d_in[17];
  const float* rb2_b2   = (const float*)d_in[18];
  const float* att_ln_g = (const float*)d_in[19];
  const float* att_ln_b = (const float*)d_in[20];
  const float* att_win  = (const float*)d_in[21];
  const float* att_bin  = (const float*)d_in[22];
  const float* att_wout = (const float*)d_in[23];
  const float* att_bout = (const float*)d_in[24];
  const float* out_w    = (const float*)d_in[25];
  const float* out_b    = (const float*)d_in[26];
  const float* fin_w    = (const float*)d_in[27];
  const float* fin_b    = (const float*)d_in[28];
  // d_in[29] = temp: tau = |temp| > 0 never changes the argmax -> unused.

  float* logits_ws = (float*)d_ws;  // 32 floats
  float* adj = (float*)d_out;       // [16][1024][1024]

  decoder_phaseA<<<1, 512, 0, stream>>>(
      x, stats, ln0_g, ln0_b, rb1_ln_g, rb1_ln_b, rb1_w1, rb1_b1, rb1_w2,
      rb1_b2, rb1_wp, rb1_bp, rb2_ln_g, rb2_ln_b, rb2_w1, rb2_b1, rb2_w2,
      rb2_b2, att_ln_g, att_ln_b, att_win, att_bin, att_wout, att_bout, out_w,
      out_b, fin_w, fin_b, logits_ws);

  decoder_pairs<<<dim3(NN, NB), 256, 0, stream>>>(u, logits_ws, adj);
}